// LocalMelSpecDiscretizer_16286515987022
// MI455X (gfx1250) — compile-verified
//
#include <hip/hip_runtime.h>
#include <hip/hip_bf16.h>

// LocalMelSpecDiscretizer for MI455X (gfx1250, wave32).
// Pure HBM-streaming kernel: 168 MB traffic -> ~7.2 us floor @ 23.3 TB/s.
// Strategy: b128 NT loads/stores of melspecs/out, centroid table (1280 B)
// broadcast into LDS via CDNA5 async-to-LDS path, argmin in VALU.
// All indexing 32-bit (n4 = 5.24M << 2^32) to avoid u64 VALU chains.

typedef float v4f __attribute__((ext_vector_type(4)));
typedef int   v4i __attribute__((ext_vector_type(4)));

typedef __attribute__((address_space(1))) v4i* gptr_b128;  // global b128
typedef __attribute__((address_space(3))) v4i* lptr_b128;  // LDS b128

#define NCH 80   // channels (C)
#define NK  4    // centroids per channel (K)

__global__ __launch_bounds__(256) void
melspec_discretize_kernel(const float* __restrict__ mel,
                          const float* __restrict__ cent,
                          float* __restrict__ out,
                          unsigned n4)
{
    // Per-channel centroid rows: sc[c] = {centroids[c][0..3]} (16 B each).
    __shared__ v4f sc[NCH];

    const unsigned tid = threadIdx.x;

#if __has_builtin(__builtin_amdgcn_global_load_async_to_lds_b128)
    // CDNA5 async data mover path: GLOBAL_LOAD_ASYNC_TO_LDS_B128 (ASYNCcnt).
    if (tid < NCH) {
        __builtin_amdgcn_global_load_async_to_lds_b128(
            (gptr_b128)(cent + tid * NK),
            (lptr_b128)&sc[tid],
            /*imm offset*/ 0, /*cpol*/ 0);
    }
  #if __has_builtin(__builtin_amdgcn_s_wait_asynccnt)
    __builtin_amdgcn_s_wait_asynccnt(0);
  #else
    asm volatile("s_wait_asynccnt 0" ::: "memory");
  #endif
#else
    // Fallback: plain coalesced b128 load + ds_store_b128.
    if (tid < NCH) {
        sc[tid] = ((const v4f*)cent)[tid];
    }
#endif
    __syncthreads();

    unsigned i = blockIdx.x * 256u + tid;   // float4 index, fits in 32 bits
    if (i >= n4) return;

    // Streaming read-once: non-temporal b128 load (th:TH_LOAD_NT).
    v4f x = __builtin_nontemporal_load((const v4f*)mel + i);

    // First flat element of this float4 is index 4*i; channel = (4*i) % 80.
    // Since 80 % 4 == 0, the 4 elements cover contiguous channels c0..c0+3.
    unsigned c0 = (i % 20u) * 4u;

    v4f r;
    #pragma unroll
    for (unsigned e = 0; e < 4; ++e) {
        v4f ce = sc[c0 + e];           // ds_load_b128: 4 centroids for channel
        float xv = x[e];
        // argmin over K=4 with first-min tie-break (strict '<' keeps lowest k),
        // matching jnp.argmin semantics; output is the winning centroid value.
        float best = ce[0];
        float bd   = __builtin_fabsf(xv - ce[0]);
        float d;
        d = __builtin_fabsf(xv - ce[1]); if (d < bd) { bd = d; best = ce[1]; }
        d = __builtin_fabsf(xv - ce[2]); if (d < bd) { bd = d; best = ce[2]; }
        d = __builtin_fabsf(xv - ce[3]); if (d < bd) { bd = d; best = ce[3]; }
        r[e] = best;
    }

    // Streaming write-once: non-temporal b128 store.
    __builtin_nontemporal_store(r, (v4f*)out + i);
}

extern "C" void kernel_launch(void* const* d_in, const int* in_sizes, int n_in,
                              void* d_out, int out_size, void* d_ws, size_t ws_size,
                              hipStream_t stream) {
    const float* mel  = (const float*)d_in[0];   // [B,T,C] f32
    const float* cent = (const float*)d_in[1];   // [C,K]   f32
    float*       out  = (float*)d_out;           // [B,T,C] f32

    unsigned n  = (unsigned)in_sizes[0];         // B*T*C (mult of 4: C=80)
    unsigned n4 = n / 4u;

    const int block = 256;  // 8 wave32 waves -> 2x WGP SIMD fill
    unsigned grid = (n4 + block - 1) / block;

    melspec_discretize_kernel<<<grid, block, 0, stream>>>(mel, cent, out, n4);
}